// PostProcessor_14955076124693
// MI455X (gfx1250) — compile-verified
//
#include <hip/hip_runtime.h>

// ---------------------------------------------------------------------------
// MI455X (gfx1250) detection post-processor: per-class threshold + sort +
// greedy NMS + global top-100.
//
// Sizing reasoning (can't measure; from specs):
//  * inputs ~3.3MB -> fully L2 resident (192MB L2). Not HBM bound.
//  * per-class working set 90KB -> fits in LDS (320KB/WGP). One WG per class.
//  * pairwise IoU ~2 GFLOP fp32 total -> latency / serial-chain bound, so the
//    NMS is tiled 16x16: serial chain is V/16 tile-resolutions, not V steps.
// CDNA5-specific paths used:
//  * TDM tensor_load_to_lds: strided gather of one class's boxes (4 floats
//    every 1296B, 2048 rows) into dense LDS, async under the score sort.
//  * V_WMMA_F32_16X16X4_F32: rank-2 outer product A=[area_r,1], B=[1;area_c]
//    computes the 16x16 tile of (area_i + area_j) IoU denominators on the
//    matrix pipe, co-executing with the VALU min/max intersection math.
//  * Inner IoU tests are branchless (predicates folded arithmetically) to
//    avoid per-pair s_and_saveexec EXEC churn seen in the previous build.
// ---------------------------------------------------------------------------

typedef float v2f __attribute__((ext_vector_type(2)));
typedef float v8f __attribute__((ext_vector_type(8)));
typedef unsigned int v4u __attribute__((ext_vector_type(4)));
typedef int v4i __attribute__((ext_vector_type(4)));
typedef int v8i __attribute__((ext_vector_type(8)));

#define N_PROP 2048
#define N_CLS  81
#define SCORE_TH 0.05f
#define NMS_TH   0.5f
#define DETS     100
#define NEGINF  (-1e9f)

__device__ __forceinline__ float clipx(float v) { return fminf(fmaxf(v, 0.0f), 1332.0f); }
__device__ __forceinline__ float clipy(float v) { return fminf(fmaxf(v, 0.0f), 799.0f); }

// 16x16 tile of pairwise area sums via one V_WMMA_F32_16X16X4_F32.
// A (16x4, ISA layout: lanes 0-15 hold K=0,1 in vgpr0,1): row m -> [ar, 1, 0, 0]
// B (4x16, lanes 0-15 hold rows K=0 (vgpr0), K=1 (vgpr1)): col n -> [1, ac, 0, 0]^T
// D[m][n] = ar[m]*1 + 1*ac[n]  (exact fp32; C=0)
// D layout: vgpr k, lanes 0-15 -> M=k, N=lane; lanes 16-31 -> M=k+8, N=lane-16.
__device__ __forceinline__ v8f area_sum_tile(const float* areav, int rb, int cb, int lane) {
    const bool lo = lane < 16;
    const float ar = areav[rb + (lane & 15)];
    const float ac = areav[cb + (lane & 15)];
    v2f a, b;
    a.x = lo ? ar : 0.0f;   a.y = lo ? 1.0f : 0.0f;
    b.x = lo ? 1.0f : 0.0f; b.y = lo ? ac : 0.0f;
    v8f c = {0.f, 0.f, 0.f, 0.f, 0.f, 0.f, 0.f, 0.f};
    return __builtin_amdgcn_wmma_f32_16x16x4_f32(false, a, false, b, (short)0, c, false, false);
}

// branchless IoU>thresh test given precomputed area-sum 'sum'
__device__ __forceinline__ unsigned iou_hit(float ax1, float ay1, float ax2, float ay2,
                                            float bx1_, float by1_, float bx2_, float by2_,
                                            float sum) {
    float xx1 = fmaxf(ax1, bx1_);
    float yy1 = fmaxf(ay1, by1_);
    float xx2 = fminf(ax2, bx2_);
    float yy2 = fminf(ay2, by2_);
    float iw = fmaxf(xx2 - xx1, 0.0f);
    float ih = fmaxf(yy2 - yy1, 0.0f);
    float inter = iw * ih;
    return (unsigned)(inter > NMS_TH * (sum - inter + 1e-9f));
}

// ---------------------------------------------------------------------------
// Phase A: one workgroup (256 thr = 8 wave32) per class.
// Dynamic LDS layout (bytes):
//   [0,32768)      rawbox  : float4[2048]   (TDM destination)
//   [32768,40960)  skey    : float[2048]
//   [40960,49152)  sidx    : uint [2048]
//   [49152,81920)  bx1/by1/bx2/by2 : 4 x float[2048]  (sorted, clipped SoA)
//   [81920,90112)  areav   : float[2048]
//   [90112,90368)  keepw   : uint [64]      (keep bitset)
//   [90368,90432)  supw    : uint [16]      (per-tile triangle bits)
//   [90432,90436)  Vsh     : int            (count above threshold)
// ---------------------------------------------------------------------------
__global__ __launch_bounds__(256) void nms_class_kernel(
    const float* __restrict__ boxes,   // [2048, 324]
    const float* __restrict__ scores,  // [2048, 81]
    float* __restrict__ cand_s,        // [80*2048]
    float4* __restrict__ cand_b) {     // [80*2048]
    extern __shared__ unsigned char smem[];
    float4*   rawbox = (float4*)(smem);
    float*    skey   = (float*)(smem + 32768);
    unsigned* sidx   = (unsigned*)(smem + 40960);
    float*    bx1    = (float*)(smem + 49152);
    float*    by1    = (float*)(smem + 57344);
    float*    bx2    = (float*)(smem + 65536);
    float*    by2    = (float*)(smem + 73728);
    float*    areav  = (float*)(smem + 81920);
    unsigned* keepw  = (unsigned*)(smem + 90112);
    unsigned* supw   = (unsigned*)(smem + 90368);
    int*      Vsh    = (int*)(smem + 90432);

    const int tid  = (int)threadIdx.x;
    const int lane = tid & 31;
    const int wid  = tid >> 5;
    const int cls  = (int)blockIdx.x + 1;   // skip background class 0

    // ---- kick off the TDM strided gather of this class's boxes (wave 0) ----
    if (wid == 0) {
        // low 32 bits of the flat LDS pointer == byte offset in LDS space
        unsigned lds_off = (unsigned)(unsigned long long)(void*)rawbox;
        unsigned long long ga =
            (unsigned long long)(const void*)(boxes + (unsigned long long)cls * 4ull);
        // D# group0: count=1 | lds_addr | global_addr[31:0] | addr[56:32], type=2
        v4u g0 = { 1u,
                   lds_off,
                   (unsigned)(ga & 0xFFFFFFFFull),
                   (unsigned)((ga >> 32) & 0x1FFFFFFull) | (2u << 30) };
        // D# group1: data_size=4B; tensor_dim0=4, tensor_dim1=2048;
        //            tile_dim0=4, tile_dim1=2048; tensor_dim0_stride=324
        v8i g1 = { 0x00020000,            // wg_mask=0, data_size=2 (4B)
                   (int)(4u << 16),       // tensor_dim0 lo16 in [31:16]
                   (int)(2048u << 16),    // tensor_dim1 lo16 in [31:16]
                   (int)(4u << 16),       // tile_dim0 in [31:16]
                   0x00000800,            // tile_dim1 = 2048
                   324,                   // tensor_dim0_stride lo32
                   0, 0 };
        v4i gz4 = { 0, 0, 0, 0 };
        v8i gz8 = { 0, 0, 0, 0, 0, 0, 0, 0 };
        // amdgpu-toolchain (clang-23) 6-arg form: g0, g1, g2, g3, g4, cpol
        __builtin_amdgcn_tensor_load_to_lds(g0, g1, gz4, gz4, gz8, 0);
    }

    // ---- load scores (strided), threshold, count survivors ----
    if (tid == 0) *Vsh = 0;
    __syncthreads();
    int lc = 0;
    for (int i = tid; i < N_PROP; i += 256) {
        float s = scores[(unsigned)i * N_CLS + cls];
        if (i + 256 < N_PROP)
            __builtin_prefetch(&scores[(unsigned)(i + 256) * N_CLS + cls], 0, 1);
        bool pass = s > SCORE_TH;
        skey[i] = pass ? s : NEGINF;
        sidx[i] = (unsigned)i;
        lc += pass ? 1 : 0;
    }
    atomicAdd(Vsh, lc);
    __syncthreads();

    // ---- bitonic sort 2048 (score desc, payload = original index) ----
    for (int k = 2; k <= N_PROP; k <<= 1) {
        for (int j = k >> 1; j > 0; j >>= 1) {
            for (int i = tid; i < N_PROP; i += 256) {
                int p = i ^ j;
                if (p > i) {
                    bool up = ((i & k) == 0);           // descending segment
                    float a = skey[i], b = skey[p];
                    bool sw = up ? (a < b) : (a > b);
                    if (sw) {
                        skey[i] = b; skey[p] = a;
                        unsigned t = sidx[i]; sidx[i] = sidx[p]; sidx[p] = t;
                    }
                }
            }
            __syncthreads();
        }
    }

    const int V = *Vsh;  // boxes 0..V-1 are above threshold after the sort

    // ---- init keep bitset: bit i set iff i < V ----
    if (tid < 64) {
        int lo = tid * 32;
        unsigned w;
        if (V >= lo + 32)      w = 0xFFFFFFFFu;
        else if (V <= lo)      w = 0u;
        else                   w = (1u << (V - lo)) - 1u;
        keepw[tid] = w;
    }
    if (wid == 0) __builtin_amdgcn_s_wait_tensorcnt(0);  // TDM tile landed
    __syncthreads();

    // ---- build sorted, clipped SoA boxes + areas ----
    for (int i = tid; i < N_PROP; i += 256) {
        float4 rb4 = rawbox[sidx[i]];
        float x1 = clipx(rb4.x), y1 = clipy(rb4.y);
        float x2 = clipx(rb4.z), y2 = clipy(rb4.w);
        bx1[i] = x1; by1[i] = y1; bx2[i] = x2; by2[i] = y2;
        areav[i] = fmaxf(x2 - x1, 0.0f) * fmaxf(y2 - y1, 0.0f);
    }
    __syncthreads();

    // ---- tiled greedy NMS: serial chain = V/16 tile resolutions ----
    const int nT = (V + 15) >> 4;
    for (int rt = 0; rt < nT; ++rt) {
        const int rb = rt << 4;
        const int half = (lane >> 4) << 3;

        // (a) wave0: raw intra-tile suppression bits (WMMA area sums).
        // All LDS indices rb..rb+15 <= 2047, so loads are unconditional and
        // the (n>m, gn<V) guards fold into the predicate.
        if (wid == 0) {
            if (lane < 16) supw[lane] = 0u;
            v8f d = area_sum_tile(areav, rb, rb, lane);
            const int n = lane & 15;
            const int gn = rb + n;
            const unsigned colok = (unsigned)(gn < V);
            const float cx1 = bx1[gn], cy1 = by1[gn];
            const float cx2 = bx2[gn], cy2 = by2[gn];
#pragma unroll
            for (int k2 = 0; k2 < 8; ++k2) {
                const int m = k2 + half;
                const int gm = rb + m;
                unsigned sup = colok & (unsigned)(n > m) &
                               iou_hit(bx1[gm], by1[gm], bx2[gm], by2[gm],
                                       cx1, cy1, cx2, cy2, d[k2]);
                if (sup) atomicOr(&supw[m], 1u << n);   // rare -> execz skip
            }
        }
        __syncthreads();

        // (b) one thread: resolve the 16x16 greedy triangle
        if (tid == 0) {
            int rows = V - rb; if (rows > 16) rows = 16;
            unsigned widx = (unsigned)rb >> 5, sh = (unsigned)rb & 31u;
            unsigned w = keepw[widx];
            unsigned cur = (w >> sh) & 0xFFFFu;
            for (int a = 0; a < rows; ++a)
                if ((cur >> a) & 1u) cur &= ~supw[a];
            keepw[widx] = (w & ~(0xFFFFu << sh)) | (cur << sh);
        }
        __syncthreads();

        // (c) all 8 waves sweep later column tiles with the kept rows
        const unsigned rowmask =
            (keepw[(unsigned)rb >> 5] >> ((unsigned)rb & 31u)) & 0xFFFFu;
        if (rowmask) {
            float rx1[8], ry1[8], rx2[8], ry2[8];
#pragma unroll
            for (int k2 = 0; k2 < 8; ++k2) {
                int gm = rb + half + k2;
                rx1[k2] = bx1[gm]; ry1[k2] = by1[gm];
                rx2[k2] = bx2[gm]; ry2[k2] = by2[gm];
            }
            for (int cb = rb + 16 + (wid << 4); cb < V; cb += 16 * 8) {
                v8f d = area_sum_tile(areav, rb, cb, lane);  // EXEC all-1s here
                const int gn = cb + (lane & 15);             // <= 2047 always
                const float cx1 = bx1[gn], cy1 = by1[gn];
                const float cx2 = bx2[gn], cy2 = by2[gn];
                unsigned kill = 0u;
#pragma unroll
                for (int k2 = 0; k2 < 8; ++k2) {
                    unsigned on = (rowmask >> (k2 + half)) & 1u;
                    kill |= on & iou_hit(rx1[k2], ry1[k2], rx2[k2], ry2[k2],
                                         cx1, cy1, cx2, cy2, d[k2]);
                }
                kill &= (unsigned)(gn < V);
                if (kill)                                    // rare -> execz skip
                    atomicAnd(&keepw[(unsigned)gn >> 5], ~(1u << ((unsigned)gn & 31u)));
            }
        }
        __syncthreads();
    }

    // ---- emit per-class candidates ----
    const unsigned base = (unsigned)blockIdx.x * N_PROP;
    for (int i = tid; i < N_PROP; i += 256) {
        bool kept = (i < V) && ((keepw[(unsigned)i >> 5] >> ((unsigned)i & 31u)) & 1u);
        cand_s[base + i] = kept ? skey[i] : NEGINF;
        cand_b[base + i] = make_float4(bx1[i], by1[i], bx2[i], by2[i]);
    }
}

// ---------------------------------------------------------------------------
// Phase B: exact top-100 over 80*2048 candidates (argmax iteration with
// flat-index tiebreak == jax.lax.top_k stable tie order). Single WG; the whole
// candidate array is L2-resident, so 100 sweeps is ~64MB of L2 traffic.
// ---------------------------------------------------------------------------
__global__ __launch_bounds__(1024) void topk_kernel(
    float* __restrict__ cand_s, const float4* __restrict__ cand_b,
    float* __restrict__ out) {
    __shared__ float rv[1024];
    __shared__ int   ri[1024];
    const int tid = (int)threadIdx.x;
    const int TOT = 80 * N_PROP;
    for (int k = 0; k < DETS; ++k) {
        float bv = -3e9f; int bi = TOT - 1;
        for (int f = tid; f < TOT; f += 1024) {
            float s = cand_s[f];
            if (s > bv) { bv = s; bi = f; }   // strided -> first hit = min index
        }
        rv[tid] = bv; ri[tid] = bi;
        __syncthreads();
        for (int off = 512; off > 0; off >>= 1) {
            if (tid < off) {
                float ov = rv[tid + off]; int oi = ri[tid + off];
                if (ov > rv[tid] || (ov == rv[tid] && oi < ri[tid])) {
                    rv[tid] = ov; ri[tid] = oi;
                }
            }
            __syncthreads();
        }
        if (tid == 0) {
            int b = ri[0];
            float4 bb = cand_b[b];
            out[k * 6 + 0] = bb.x;
            out[k * 6 + 1] = bb.y;
            out[k * 6 + 2] = bb.z;
            out[k * 6 + 3] = bb.w;
            out[k * 6 + 4] = rv[0];
            out[k * 6 + 5] = (float)((b >> 11) + 1);  // class label
            cand_s[b] = -2e9f;                        // exclude from next sweep
        }
        __syncthreads();
        __threadfence();
    }
}

extern "C" void kernel_launch(void* const* d_in, const int* in_sizes, int n_in,
                              void* d_out, int out_size, void* d_ws, size_t ws_size,
                              hipStream_t stream) {
    const float* boxes  = (const float*)d_in[0];   // [2048*324] f32
    const float* scores = (const float*)d_in[1];   // [2048*81]  f32
    float* out = (float*)d_out;                    // [100*6]    f32

    // workspace: cand scores (640KB) then cand boxes (2.5MB) -> 3.25MB total
    float*  cand_s = (float*)d_ws;
    float4* cand_b = (float4*)((char*)d_ws + (size_t)80 * N_PROP * sizeof(float));

    (void)in_sizes; (void)n_in; (void)out_size; (void)ws_size;

    nms_class_kernel<<<80, 256, 90496, stream>>>(boxes, scores, cand_s, cand_b);
    topk_kernel<<<1, 1024, 0, stream>>>(cand_s, cand_b, out);
}